// RoiHead_32212254720115
// MI455X (gfx1250) — compile-verified
//
#include <hip/hip_runtime.h>

typedef __attribute__((ext_vector_type(2))) float v2f;
typedef __attribute__((ext_vector_type(8))) float v8f;

#define B_DIM   16
#define C_CH    528
#define T_DIM   32
#define H_DIM   28
#define W_DIM   28
#define HW      (H_DIM * W_DIM)        // 784
#define R_ROI   256
#define N1      128
#define N2      32

// workspace layout (in floats)
#define FEAT_N  (B_DIM * C_CH * HW)    // 6,623,232
#define R_N     (R_ROI * C_CH)         // 135,168
#define H1_N    (R_ROI * N1)           // 32,768
#define H2_N    (R_ROI * N2)           // 8,192

// ---------------------------------------------------------------------------
// Kernel 1: temporal mean over T=32.  Pure HBM streaming (~848 MB read),
// fully coalesced float4 loads.  Total float4 outputs = 16*528*196 = 1,655,808
// = 6468 blocks * 256 threads exactly (no bounds check needed).
// ---------------------------------------------------------------------------
__global__ void tmean_kernel(const float* __restrict__ x, float* __restrict__ feat) {
    const int gid   = blockIdx.x * blockDim.x + threadIdx.x;   // float4 index
    const int cflat = gid / (HW / 4);                          // b*528 + c
    const int q     = gid - cflat * (HW / 4);                  // float4 within hw
    const float4* xp = (const float4*)x;
    const size_t base = (size_t)cflat * (T_DIM * (HW / 4)) + q;
    float ax = 0.f, ay = 0.f, az = 0.f, aw = 0.f;
    #pragma unroll
    for (int t = 0; t < T_DIM; ++t) {
        float4 v = xp[base + (size_t)t * (HW / 4)];
        ax += v.x; ay += v.y; az += v.z; aw += v.w;
    }
    float4 o;
    o.x = ax * (1.f / T_DIM); o.y = ay * (1.f / T_DIM);
    o.z = az * (1.f / T_DIM); o.w = aw * (1.f / T_DIM);
    ((float4*)feat)[gid] = o;
}

// ---------------------------------------------------------------------------
// Kernel 2: ROI-align + full spatial mean -> r[256][528].
// One block per ROI.  The 256 sample points' bilinear taps (4 offsets + 4
// weights, weights folded with validity and the 1/256 average) are channel-
// invariant: compute once into 8 KB LDS, then one thread per channel streams
// them (uniform-address LDS broadcast) while gathering from the 3 KB per-
// channel feat tile (resident in L2/L1).
// ---------------------------------------------------------------------------
__global__ void roi_pool_kernel(const float* __restrict__ feat,
                                const float* __restrict__ bbox,
                                float* __restrict__ rout) {
    __shared__ int   s_off[256][4];
    __shared__ float s_w[256][4];

    const int roi = blockIdx.x;
    const int t   = threadIdx.x;

    const float bf = bbox[roi * 5 + 0];
    const float x1 = bbox[roi * 5 + 1] * (1.f / 16.f) - 0.5f;
    const float y1 = bbox[roi * 5 + 2] * (1.f / 16.f) - 0.5f;
    const float x2 = bbox[roi * 5 + 3] * (1.f / 16.f) - 0.5f;
    const float y2 = bbox[roi * 5 + 4] * (1.f / 16.f) - 0.5f;
    const int   bi = (int)bf;
    const float bin_w = (x2 - x1) * 0.125f;   // /OUT
    const float bin_h = (y2 - y1) * 0.125f;

    // sample point for this thread: i = y-index, j = x-index (S = 16)
    const int i = t >> 4;
    const int j = t & 15;
    const float yy = y1 + ((float)i + 0.5f) * 0.5f * bin_h;
    const float xx = x1 + ((float)j + 0.5f) * 0.5f * bin_w;
    const bool valid = (yy > -1.0f) && (yy < (float)H_DIM) &&
                       (xx > -1.0f) && (xx < (float)W_DIM);
    const float yc = fminf(fmaxf(yy, 0.f), (float)(H_DIM - 1));
    const float xc = fminf(fmaxf(xx, 0.f), (float)(W_DIM - 1));
    const int y0 = (int)floorf(yc);
    const int x0 = (int)floorf(xc);
    const int y1i = min(y0 + 1, H_DIM - 1);
    const int x1i = min(x0 + 1, W_DIM - 1);
    const float ly = yc - (float)y0;
    const float lx = xc - (float)x0;
    const float hy = 1.f - ly;
    const float hx = 1.f - lx;
    const float vf = valid ? (1.f / 256.f) : 0.f;  // fold validity + mean

    s_off[t][0] = y0 * W_DIM + x0;
    s_off[t][1] = y0 * W_DIM + x1i;
    s_off[t][2] = y1i * W_DIM + x0;
    s_off[t][3] = y1i * W_DIM + x1i;
    s_w[t][0] = hy * hx * vf;
    s_w[t][1] = hy * lx * vf;
    s_w[t][2] = ly * hx * vf;
    s_w[t][3] = ly * lx * vf;
    __syncthreads();

    const float* fb = feat + (size_t)bi * C_CH * HW;
    for (int c = t; c < C_CH; c += 256) {
        const float* fc = fb + (size_t)c * HW;
        float acc = 0.f;
        #pragma unroll 4
        for (int s = 0; s < 256; ++s) {
            acc += s_w[s][0] * fc[s_off[s][0]]
                 + s_w[s][1] * fc[s_off[s][1]]
                 + s_w[s][2] * fc[s_off[s][2]]
                 + s_w[s][3] * fc[s_off[s][3]];
        }
        rout[roi * C_CH + c] = acc;
    }
}

// ---------------------------------------------------------------------------
// Kernel 3: h1 = relu(r @ w1^T + b1)   (256x528)(528x128) -> 256x128
// One wave per 16x16 output tile, V_WMMA_F32_16X16X4_F32 over K.
// Fragment layout (wave32): lane l, half = l>>4, rr = l&15.
//   A[16x4]: a[j] = A[row][kb + 2*half + j]
//   B[4x16]: b[j] = B[kb + 2*half + j][col]   (B[k][n] = w1[n*528 + k])
//   D:       acc[v] = D[mt*16 + v + 8*half][col]
// ---------------------------------------------------------------------------
__global__ void gemm1_kernel(const float* __restrict__ A,
                             const float* __restrict__ W1,
                             const float* __restrict__ bias,
                             float* __restrict__ out) {
    const int lane = threadIdx.x & 31;
    const int wv   = threadIdx.x >> 5;
    const int tile = blockIdx.x * 8 + wv;      // 0..127
    const int mt   = tile >> 3;                // 0..15
    const int nt   = tile & 7;                 // 0..7
    const int half = lane >> 4;
    const int rr   = lane & 15;
    const int row  = mt * 16 + rr;
    const int col  = nt * 16 + rr;
    const float* arow = A  + (size_t)row * C_CH;
    const float* wrow = W1 + (size_t)col * C_CH;

    v8f acc = {};
    for (int kb = 0; kb < C_CH; kb += 4) {
        const int k0 = kb + 2 * half;
        v2f a; a.x = arow[k0]; a.y = arow[k0 + 1];
        v2f b; b.x = wrow[k0]; b.y = wrow[k0 + 1];
        acc = __builtin_amdgcn_wmma_f32_16x16x4_f32(
            false, a, false, b, (short)0, acc, false, false);
    }
    const float bc = bias[col];
    #pragma unroll
    for (int v = 0; v < 8; ++v) {
        const int m = mt * 16 + v + 8 * half;
        out[m * N1 + col] = fmaxf(acc[v] + bc, 0.f);
    }
}

// ---------------------------------------------------------------------------
// Kernel 4: h2 = h1 @ w2^T + b2   (256x128)(128x32) -> 256x32
// ---------------------------------------------------------------------------
__global__ void gemm2_kernel(const float* __restrict__ A,
                             const float* __restrict__ W2,
                             const float* __restrict__ bias,
                             float* __restrict__ out) {
    const int lane = threadIdx.x & 31;
    const int wv   = threadIdx.x >> 5;
    const int tile = blockIdx.x * 8 + wv;      // 0..31
    const int mt   = tile >> 1;                // 0..15
    const int nt   = tile & 1;                 // 0..1
    const int half = lane >> 4;
    const int rr   = lane & 15;
    const int row  = mt * 16 + rr;
    const int col  = nt * 16 + rr;
    const float* arow = A  + (size_t)row * N1;
    const float* wrow = W2 + (size_t)col * N1;

    v8f acc = {};
    for (int kb = 0; kb < N1; kb += 4) {
        const int k0 = kb + 2 * half;
        v2f a; a.x = arow[k0]; a.y = arow[k0 + 1];
        v2f b; b.x = wrow[k0]; b.y = wrow[k0 + 1];
        acc = __builtin_amdgcn_wmma_f32_16x16x4_f32(
            false, a, false, b, (short)0, acc, false, false);
    }
    const float bc = bias[col];
    #pragma unroll
    for (int v = 0; v < 8; ++v) {
        const int m = mt * 16 + v + 8 * half;
        out[m * N2 + col] = acc[v] + bc;
    }
}

// ---------------------------------------------------------------------------
// Kernel 5: out = h2 @ w3^T + b3   (256x32)(32x1) -> 256
// ---------------------------------------------------------------------------
__global__ void head_kernel(const float* __restrict__ h2,
                            const float* __restrict__ w3,
                            const float* __restrict__ b3,
                            float* __restrict__ out) {
    const int m = threadIdx.x;
    float acc = b3[0];
    #pragma unroll
    for (int k = 0; k < N2; ++k)
        acc += h2[m * N2 + k] * w3[k];
    out[m] = acc;
}

// ---------------------------------------------------------------------------
extern "C" void kernel_launch(void* const* d_in, const int* in_sizes, int n_in,
                              void* d_out, int out_size, void* d_ws, size_t ws_size,
                              hipStream_t stream) {
    const float* x    = (const float*)d_in[0];
    const float* bbox = (const float*)d_in[1];
    const float* w1   = (const float*)d_in[2];
    const float* b1   = (const float*)d_in[3];
    const float* w2   = (const float*)d_in[4];
    const float* b2   = (const float*)d_in[5];
    const float* w3   = (const float*)d_in[6];
    const float* b3   = (const float*)d_in[7];

    float* ws   = (float*)d_ws;
    float* feat = ws;
    float* rbuf = feat + FEAT_N;
    float* h1   = rbuf + R_N;
    float* h2   = h1 + H1_N;
    float* out  = (float*)d_out;

    // 1) temporal mean: 1,655,808 float4 outputs = 6468 * 256
    tmean_kernel<<<6468, 256, 0, stream>>>(x, feat);
    // 2) ROI align + spatial mean
    roi_pool_kernel<<<R_ROI, 256, 0, stream>>>(feat, bbox, rbuf);
    // 3) MLP layer 1 (WMMA, 128 tiles -> 16 blocks * 8 waves)
    gemm1_kernel<<<16, 256, 0, stream>>>(rbuf, w1, b1, h1);
    // 4) MLP layer 2 (WMMA, 32 tiles -> 4 blocks * 8 waves)
    gemm2_kernel<<<4, 256, 0, stream>>>(h1, w2, b2, h2);
    // 5) final projection
    head_kernel<<<1, 256, 0, stream>>>(h2, w3, b3, out);
}